// GAEAttention_8065948582032
// MI455X (gfx1250) — compile-verified
//
#include <hip/hip_runtime.h>
#include <hip/hip_bf16.h>

// ---------------------------------------------------------------------------
// GAEAttention (SE-with-trivial-GCN) for MI455X / gfx1250.
// Memory-bound: ~924 MB HBM traffic -> ~40us floor @ 23.3 TB/s.
// Tiny GEMMs done with V_WMMA_F32_16X16X4_F32 (CDNA5 fp32 WMMA, exact fp32).
// ---------------------------------------------------------------------------

typedef __attribute__((ext_vector_type(2))) float v2f;
typedef __attribute__((ext_vector_type(4))) float f4;
typedef __attribute__((ext_vector_type(8))) float v8f;

#define BATCH 32
#define CH    768
#define HID   48
#define HW    3136          // 56*56
#define HW4   784           // HW / 4 (float4 per plane)
#define NPLANE (BATCH * CH) // 24576

// ---------------------------------------------------------------------------
// Kernel 1: global average pool. One 256-thread block per (b,c) plane.
// Coalesced b128 loads, LDS tree reduction.
// ---------------------------------------------------------------------------
__global__ __launch_bounds__(256)
void gae_pool_kernel(const float* __restrict__ x, float* __restrict__ y) {
    __shared__ float red[256];
    const int plane = blockIdx.x;
    const f4* __restrict__ xv = (const f4*)x + (size_t)plane * HW4;

    float s = 0.0f;
    for (int i = threadIdx.x; i < HW4; i += 256) {
        f4 v = xv[i];
        s += (v.x + v.y) + (v.z + v.w);
    }
    red[threadIdx.x] = s;
    __syncthreads();
    #pragma unroll
    for (int off = 128; off > 0; off >>= 1) {
        if ((int)threadIdx.x < off) red[threadIdx.x] += red[threadIdx.x + off];
        __syncthreads();
    }
    if (threadIdx.x == 0) y[plane] = red[0] * (1.0f / (float)HW);
}

// ---------------------------------------------------------------------------
// WMMA helpers: V_WMMA_F32_16X16X4_F32, D(16x16) = A(16x4) * B(4x16) + C.
// Per ISA 7.12.2 (32-bit A 16x4): lanes 0-15 hold K={0,1}, lanes 16-31 K={2,3},
// outer index (M for A, N for B) = lane & 15. Both operands are loaded from
// row-major [outer x K] storage (B comes from W^T, i.e. W stored [N x K]),
// so a single loader covers A and B.
// C/D: VGPR r, lane L -> element (M = r + 8*(L>>4), N = L&15).
// ---------------------------------------------------------------------------
__device__ __forceinline__
v2f wmma_load_operand(const float* __restrict__ m, int ld, int outer0, int k0, int lane) {
    const int r  = outer0 + (lane & 15);
    const int kb = k0 + ((lane >> 4) << 1);
    v2f o;
    o.x = m[r * ld + kb];
    o.y = m[r * ld + kb + 1];
    return o;
}

__device__ __forceinline__
v8f wmma_f32(v2f a, v2f b, v8f c) {
    return __builtin_amdgcn_wmma_f32_16x16x4_f32(
        /*neg_a=*/false, a, /*neg_b=*/false, b,
        /*c_mod=*/(short)0, c, /*reuse_a=*/false, /*reuse_b=*/false);
}

// ---------------------------------------------------------------------------
// Kernel 2: z = relu(y @ w_fc^T) @ w_gcn^T. Single workgroup, 8 waves.
//   Stage A: h[32,48]  (M=32,N=48,K=768) -> 6 tiles, one per wave (waves 0-5)
//   Stage B: z[32,768] (M=32,N=768,K=48) -> 96 tiles, 12 per wave
// All tile ownership is wave-uniform so EXEC is all-1s around every WMMA.
// ---------------------------------------------------------------------------
__global__ __launch_bounds__(256)
void gae_gemm_kernel(const float* __restrict__ y,     // [32,768]
                     const float* __restrict__ w_fc,  // [48,768]
                     const float* __restrict__ w_gcn, // [768,48]
                     float* __restrict__ z) {         // [32,768]
    __shared__ float h_lds[BATCH * HID];   // 6 KB
    const int lane = threadIdx.x & 31;
    const int wave = threadIdx.x >> 5;     // 0..7

    // ---- Stage A: h = relu(y @ w_fc^T) ----
    if (wave < 6) {
        const int m0 = (wave & 1) * 16;    // 0,16
        const int n0 = (wave >> 1) * 16;   // 0,16,32
        v8f acc = {};
        for (int k0 = 0; k0 < CH; k0 += 4) {
            v2f a = wmma_load_operand(y,    CH, m0, k0, lane);
            v2f b = wmma_load_operand(w_fc, CH, n0, k0, lane);
            acc = wmma_f32(a, b, acc);
        }
        const int mrow = (lane >> 4) << 3;   // 0 or 8
        const int col  = n0 + (lane & 15);
        #pragma unroll
        for (int r = 0; r < 8; ++r) {
            float v = acc[r];
            h_lds[(m0 + mrow + r) * HID + col] = v > 0.0f ? v : 0.0f;
        }
    }
    __syncthreads();

    // ---- Stage B: z = h @ w_gcn^T ----
    for (int t = wave; t < 96; t += 8) {
        const int m0 = (t & 1) * 16;
        const int n0 = (t >> 1) * 16;
        v8f acc = {};
        #pragma unroll
        for (int k0 = 0; k0 < HID; k0 += 4) {
            v2f a = wmma_load_operand(h_lds, HID, m0, k0, lane);
            v2f b = wmma_load_operand(w_gcn, HID, n0, k0, lane);
            acc = wmma_f32(a, b, acc);
        }
        const int mrow = (lane >> 4) << 3;
        const int col  = n0 + (lane & 15);
        #pragma unroll
        for (int r = 0; r < 8; ++r)
            z[(m0 + mrow + r) * CH + col] = acc[r];
    }
}

// ---------------------------------------------------------------------------
// Kernel 3: out = x * z[plane]. One block per plane; uniform scale factor,
// b128 loads, non-temporal b128 stores (308 MB streamed out, keep L2 clean).
// ---------------------------------------------------------------------------
__global__ __launch_bounds__(256)
void gae_scale_kernel(const float* __restrict__ x,
                      const float* __restrict__ z,
                      float* __restrict__ out) {
    const int plane = blockIdx.x;
    const float s = z[plane];                 // uniform -> scalar load
    const f4* __restrict__ xv = (const f4*)x + (size_t)plane * HW4;
    f4* __restrict__ ov       = (f4*)out + (size_t)plane * HW4;

    for (int i = threadIdx.x; i < HW4; i += 256) {
        f4 v = xv[i];
        v.x *= s; v.y *= s; v.z *= s; v.w *= s;
        __builtin_nontemporal_store(v, ov + i);
    }
}

// ---------------------------------------------------------------------------
extern "C" void kernel_launch(void* const* d_in, const int* in_sizes, int n_in,
                              void* d_out, int out_size, void* d_ws, size_t ws_size,
                              hipStream_t stream) {
    const float* x     = (const float*)d_in[0];  // [32,768,56,56]
    const float* w_fc  = (const float*)d_in[1];  // [48,768]
    const float* w_gcn = (const float*)d_in[2];  // [768,48]
    float* out = (float*)d_out;                  // [32,768,56,56]

    float* y = (float*)d_ws;                     // [32,768]  (96 KB)
    float* z = y + (size_t)BATCH * CH;           // [32,768]  (96 KB)

    gae_pool_kernel <<<NPLANE, 256, 0, stream>>>(x, y);
    gae_gemm_kernel <<<1,      256, 0, stream>>>(y, w_fc, w_gcn, z);
    gae_scale_kernel<<<NPLANE, 256, 0, stream>>>(x, z, out);
}